// GNN_27161373179970
// MI455X (gfx1250) — compile-verified
//
#include <hip/hip_runtime.h>

typedef __attribute__((ext_vector_type(2))) float v2f;
typedef __attribute__((ext_vector_type(8))) float v8f;

#define N_NODES 50000
#define N_EDGES 800000
#define IN_DIM  128
#define H_HEADS 4
#define OUT_DIM 16
#define EMB_DIM 32
#define EF_DIM  8
#define HO      (H_HEADS * OUT_DIM)   // 64
#define HE      (H_HEADS * EMB_DIM)   // 128

// ---------------------------------------------------------------------------
// out[M,P] = X[M,128] @ W[128,P], exact fp32 via V_WMMA_F32_16X16X4_F32.
// One wave per 16x16 output tile; K-loop of 32 WMMAs.
// A frag (16x4 f32, 2 VGPR): lane<16 -> row=m0+lane, K={k0,k0+1};
//                            lane>=16 -> row=m0+lane-16, K={k0+2,k0+3}.
// B frag (4x16 f32, 2 VGPR): lane<16 -> col=n0+lane, rows {k0,k0+1};
//                            lane>=16 -> col=n0+lane-16, rows {k0+2,k0+3}.
// D (16x16 f32, 8 VGPR): acc[v] = D[m0+v(+8 for hi lanes)][col].
// ---------------------------------------------------------------------------
__global__ void gemm_f32_wmma(const float* __restrict__ X,
                              const float* __restrict__ W,
                              float* __restrict__ out, int P) {
  const int m0   = blockIdx.x * 16;
  const int n0   = blockIdx.y * 16;
  const int lane = threadIdx.x & 31;
  const bool hi  = lane >= 16;
  const int l16  = lane & 15;
  const int arow = m0 + l16;
  const int koff = hi ? 2 : 0;
  const int bcol = n0 + l16;

  v8f acc = {};
  for (int k0 = 0; k0 < IN_DIM; k0 += 4) {
    v2f a = *(const v2f*)(X + (size_t)arow * IN_DIM + k0 + koff);
    v2f b;
    b.x = W[(size_t)(k0 + koff + 0) * P + bcol];
    b.y = W[(size_t)(k0 + koff + 1) * P + bcol];
    acc = __builtin_amdgcn_wmma_f32_16x16x4_f32(
        /*neg_a=*/false, a, /*neg_b=*/false, b,
        /*c_mod=*/(short)0, acc, /*reuse_a=*/false, /*reuse_b=*/false);
  }

  const int rbase = m0 + (hi ? 8 : 0);
#pragma unroll
  for (int v = 0; v < 8; ++v)
    out[(size_t)(rbase + v) * P + bcol] = acc[v];
}

// ---------------------------------------------------------------------------
// d_out := bias (broadcast), denom := 0. Fully rewritten every call.
// ---------------------------------------------------------------------------
__global__ void init_out(float* __restrict__ rst, float* __restrict__ denom,
                         const float* __restrict__ bias) {
  const int i = blockIdx.x * blockDim.x + threadIdx.x;
  if (i < N_NODES * HO)      rst[i]   = bias[i & (HO - 1)];
  if (i < N_NODES * H_HEADS) denom[i] = 0.0f;
}

// ---------------------------------------------------------------------------
// One wave per edge; lane d = embedding dim. For each head h:
//   ek = bek + bond[e] @ Wek   (8 MACs, Wek column h*32+d)
//   p  = q[src][h,d] * k[dst][h,d] * ek ;  wave-sum(p) -> logit
//   es[e,h] = exp(logit);  denom[dst,h] += es  (atomic, L2)
// Max-subtraction omitted: exp(e)/sum(exp(e)) is exact softmax, |e| is small.
// ---------------------------------------------------------------------------
__global__ void edge_logits(const float* __restrict__ q,
                            const float* __restrict__ k,
                            const float* __restrict__ bond,
                            const int* __restrict__ src,
                            const int* __restrict__ dst,
                            const float* __restrict__ Wek,
                            const float* __restrict__ bek,
                            float* __restrict__ es,
                            float* __restrict__ denom) {
  const int wave = threadIdx.x >> 5;
  const int lane = threadIdx.x & 31;
  const int e    = blockIdx.x * (blockDim.x >> 5) + wave;
  if (e >= N_EDGES) return;

  const int s = src[e];
  const int t = dst[e];

  float bf[EF_DIM];
#pragma unroll
  for (int f = 0; f < EF_DIM; ++f) bf[f] = bond[(size_t)e * EF_DIM + f];

#pragma unroll
  for (int h = 0; h < H_HEADS; ++h) {
    const int idx = h * EMB_DIM + lane;
    float ek = bek[idx];
#pragma unroll
    for (int f = 0; f < EF_DIM; ++f)
      ek = fmaf(bf[f], Wek[f * HE + idx], ek);
    float p = q[(size_t)s * HE + idx] * k[(size_t)t * HE + idx] * ek;
#pragma unroll
    for (int m = 16; m >= 1; m >>= 1)
      p += __shfl_xor(p, m, 32);
    if (lane == 0) {
      const float ex = expf(p);
      es[(size_t)e * H_HEADS + h] = ex;
      atomicAdd(&denom[(size_t)t * H_HEADS + h], ex);
    }
  }
}

// ---------------------------------------------------------------------------
// One wave per edge; lane covers (h,o) slots j = lane and lane+32 (HO=64):
//   ef  = bef + bond[e] @ Wef
//   a   = es[e,h] / denom[dst,h]
//   out[dst, j] += ft[src, j] * ef * a      (atomic, L2)
// ---------------------------------------------------------------------------
__global__ void edge_message(const float* __restrict__ ft,
                             const float* __restrict__ bond,
                             const int* __restrict__ src,
                             const int* __restrict__ dst,
                             const float* __restrict__ Wef,
                             const float* __restrict__ bef,
                             const float* __restrict__ es,
                             const float* __restrict__ denom,
                             float* __restrict__ rst) {
  const int wave = threadIdx.x >> 5;
  const int lane = threadIdx.x & 31;
  const int e    = blockIdx.x * (blockDim.x >> 5) + wave;
  if (e >= N_EDGES) return;

  const int s = src[e];
  const int t = dst[e];

  float bf[EF_DIM];
#pragma unroll
  for (int f = 0; f < EF_DIM; ++f) bf[f] = bond[(size_t)e * EF_DIM + f];

#pragma unroll
  for (int half = 0; half < 2; ++half) {
    const int j = lane + half * 32;     // 0..63 -> (h = j/16, o = j%16)
    const int h = j >> 4;
    float ef = bef[j];
#pragma unroll
    for (int f = 0; f < EF_DIM; ++f)
      ef = fmaf(bf[f], Wef[f * HO + j], ef);
    const float a   = es[(size_t)e * H_HEADS + h] / denom[(size_t)t * H_HEADS + h];
    const float msg = ft[(size_t)s * HO + j] * ef * a;
    atomicAdd(&rst[(size_t)t * HO + j], msg);
  }
}

// ---------------------------------------------------------------------------
extern "C" void kernel_launch(void* const* d_in, const int* in_sizes, int n_in,
                              void* d_out, int out_size, void* d_ws, size_t ws_size,
                              hipStream_t stream) {
  (void)in_sizes; (void)n_in; (void)out_size; (void)ws_size;

  const float* feat = (const float*)d_in[0];
  const float* bond = (const float*)d_in[1];
  const int*   src  = (const int*)d_in[2];
  const int*   dst  = (const int*)d_in[3];
  const float* Wq   = (const float*)d_in[4];
  const float* Wk   = (const float*)d_in[5];
  const float* Wv   = (const float*)d_in[6];
  const float* Wek  = (const float*)d_in[7];
  const float* bek  = (const float*)d_in[8];
  const float* Wef  = (const float*)d_in[9];
  const float* bef  = (const float*)d_in[10];
  const float* bias = (const float*)d_in[11];

  float* out = (float*)d_out;

  // Workspace layout (floats): q | k | ft | es | denom  (~78 MB total)
  float* ws    = (float*)d_ws;
  float* q     = ws;
  float* kk    = q  + (size_t)N_NODES * HE;
  float* ft    = kk + (size_t)N_NODES * HE;
  float* es    = ft + (size_t)N_NODES * HO;
  float* denom = es + (size_t)N_EDGES * H_HEADS;

  // 1) init output (bias) + denom (zero)
  init_out<<<(N_NODES * HO + 255) / 256, 256, 0, stream>>>(out, denom, bias);

  // 2) node projections via fp32 WMMA (N = 3125 * 16 exactly)
  dim3 gq(N_NODES / 16, HE / 16);   // q, k : 3125 x 8 tiles
  gemm_f32_wmma<<<gq, 32, 0, stream>>>(feat, Wq, q,  HE);
  gemm_f32_wmma<<<gq, 32, 0, stream>>>(feat, Wk, kk, HE);
  dim3 gv(N_NODES / 16, HO / 16);   // ft  : 3125 x 4 tiles
  gemm_f32_wmma<<<gv, 32, 0, stream>>>(feat, Wv, ft, HO);

  // 3) edge logits + segment-softmax denominator (8 edges per 256-thread block)
  const int eblocks = (N_EDGES + 7) / 8;
  edge_logits<<<eblocks, 256, 0, stream>>>(q, kk, bond, src, dst, Wek, bek, es, denom);

  // 4) weighted message scatter-aggregate into d_out
  edge_message<<<eblocks, 256, 0, stream>>>(ft, bond, src, dst, Wef, bef, es, denom, out);
}